// CenterLoss_60885456388837
// MI455X (gfx1250) — compile-verified
//
#include <hip/hip_runtime.h>

// CenterLoss forward+backward for MI455X (gfx1250, wave32).
// d_in[0] = y (int, BATCH), d_in[1] = feat (f32, BATCH x 128),
// d_in[2] = centers (f32, 100000 x 128)
// d_out[0] = loss, d_out[1..] = centers_grad (100000 x 128), row major.

#define NUM_CLASSES 100000
#define FEAT_DIM    128
#define BATCH       16384
#define HALF_LOSS_W 0.005f   // LOSS_WEIGHT * 0.5

typedef __attribute__((ext_vector_type(2))) float v2f;
typedef __attribute__((ext_vector_type(8))) float v8f;

__device__ __forceinline__ void atom_add_f32(float* p, float v) {
    // no-return float atomic -> global_atomic_add_f32 (STOREcnt path)
    (void)__hip_atomic_fetch_add(p, v, __ATOMIC_RELAXED, __HIP_MEMORY_SCOPE_AGENT);
}

// ---------------------------------------------------------------------------
// Kernel 1: zero counts[NUM_CLASSES] + feat_sum[NUM_CLASSES*FEAT_DIM] (+loss).
// 12,900,000 floats = 3,225,000 float4 stores, grid-stride, b128 stores.
// ---------------------------------------------------------------------------
__global__ void cl_init_kernel(float* __restrict__ ws, float* __restrict__ out_loss) {
    const size_t total4 = (size_t)NUM_CLASSES * (FEAT_DIM + 1) / 4; // 3,225,000
    float4* p = (float4*)ws;
    size_t idx    = (size_t)blockIdx.x * blockDim.x + threadIdx.x;
    size_t stride = (size_t)gridDim.x * blockDim.x;
    const float4 z = make_float4(0.f, 0.f, 0.f, 0.f);
    for (size_t i = idx; i < total4; i += stride) p[i] = z;
    if (idx == 0) *out_loss = 0.f;
}

// ---------------------------------------------------------------------------
// Kernel 2: one wave32 per sample (8 samples / 256-thread block).
// Lane l handles features [4l, 4l+4): float4 load of feat & gathered center,
// diff^2 partial, float-atomic scatter into feat_sum, count atomic.
// Wave loss reduction via V_WMMA_F32_16X16X4_F32 ones-matmul trick.
// ---------------------------------------------------------------------------
__global__ void cl_scatter_loss_kernel(const int* __restrict__ y,
                                       const float* __restrict__ feat,
                                       const float* __restrict__ centers,
                                       float* __restrict__ counts,
                                       float* __restrict__ fsum,
                                       float* __restrict__ out_loss) {
    const int lane = threadIdx.x & 31;
    const int wave = threadIdx.x >> 5;
    const int i    = blockIdx.x * 8 + wave;      // sample index, always < BATCH
    const int c    = y[i];

    const float4 f  = *(const float4*)(feat    + (size_t)i * FEAT_DIM + lane * 4);
    const float4 ce = *(const float4*)(centers + (size_t)c * FEAT_DIM + lane * 4);

    const float dx = f.x - ce.x, dy = f.y - ce.y, dz = f.z - ce.z, dw = f.w - ce.w;
    const float p  = dx * dx + dy * dy + dz * dz + dw * dw;

    // scatter-add feature sums (mostly uncontended: <=16384 hits over 100000 rows)
    float* fs = fsum + (size_t)c * FEAT_DIM + lane * 4;
    atom_add_f32(fs + 0, f.x);
    atom_add_f32(fs + 1, f.y);
    atom_add_f32(fs + 2, f.z);
    atom_add_f32(fs + 3, f.w);
    if (lane == 0) atom_add_f32(counts + c, 1.0f);

    // ---- wave reduction of p on the matrix pipe ----
    // A (16x4 f32): VGPR0 holds K0 (lanes 0-15) / K2 (lanes 16-31) = p;
    // VGPR1 (K1/K3) = 0. B = all ones => D[m][n] = p_m + p_{m+16}.
    v2f a; a.x = p;   a.y = 0.0f;
    v2f b; b.x = 1.0f; b.y = 1.0f;
    v8f acc = {};
    acc = __builtin_amdgcn_wmma_f32_16x16x4_f32(
        /*neg_a=*/false, a, /*neg_b=*/false, b,
        /*c_mod=*/(short)0, acc, /*reuse_a=*/false, /*reuse_b=*/false);
    // lanes 0-15 hold rowsums M=0..7 in acc[0..7]; lanes 16-31 hold M=8..15
    float s = acc[0] + acc[1] + acc[2] + acc[3] + acc[4] + acc[5] + acc[6] + acc[7];
    s += __shfl_xor(s, 16, 32);                  // full 32-lane sum in every lane

    __shared__ float wpart[8];
    if (lane == 0) wpart[wave] = s;
    __syncthreads();
    if (threadIdx.x == 0) {
        float t = 0.f;
#pragma unroll
        for (int w = 0; w < 8; ++w) t += wpart[w];
        atom_add_f32(out_loss, HALF_LOSS_W * t);
    }
}

// ---------------------------------------------------------------------------
// Kernel 3: one wave32 per class (8 classes / block). Empty class => grad row
// is exactly zero (ratio = 0): skip centers/feat_sum reads entirely (~84% of
// rows). Output base is d_out+1 (4B-aligned only) -> 4x b32 stores per lane.
// ---------------------------------------------------------------------------
__global__ void cl_grad_kernel(const float* __restrict__ centers,
                               const float* __restrict__ counts,
                               const float* __restrict__ fsum,
                               float* __restrict__ grad) {
    const int lane = threadIdx.x & 31;
    const int wave = threadIdx.x >> 5;
    const int c    = blockIdx.x * 8 + wave;      // class index, always < NUM_CLASSES

    const float cnt = counts[c];                 // uniform per wave
    float* g = grad + (size_t)c * FEAT_DIM + lane * 4;

    if (cnt == 0.0f) {
        g[0] = 0.f; g[1] = 0.f; g[2] = 0.f; g[3] = 0.f;
        return;
    }
    const float ratio = cnt / (1.0f + cnt);
    const float inv   = 1.0f / cnt;              // cnt >= 1 here, matches max(cnt,1)

    const float4 ce = *(const float4*)(centers + (size_t)c * FEAT_DIM + lane * 4);
    const float4 fs = *(const float4*)(fsum    + (size_t)c * FEAT_DIM + lane * 4);

    g[0] = ratio * (ce.x - fs.x * inv);
    g[1] = ratio * (ce.y - fs.y * inv);
    g[2] = ratio * (ce.z - fs.z * inv);
    g[3] = ratio * (ce.w - fs.w * inv);
}

// ---------------------------------------------------------------------------
extern "C" void kernel_launch(void* const* d_in, const int* in_sizes, int n_in,
                              void* d_out, int out_size, void* d_ws, size_t ws_size,
                              hipStream_t stream) {
    const int*   y       = (const int*)d_in[0];     // per harness: integer -> int*
    const float* feat    = (const float*)d_in[1];
    const float* centers = (const float*)d_in[2];
    float*       out     = (float*)d_out;           // out[0]=loss, out[1..]=grad

    float* counts = (float*)d_ws;                   // NUM_CLASSES floats
    float* fsum   = counts + NUM_CLASSES;           // NUM_CLASSES*FEAT_DIM floats
                                                    // (offset 400000 B, 16B aligned)

    cl_init_kernel<<<2048, 256, 0, stream>>>(counts, out);
    cl_scatter_loss_kernel<<<BATCH / 8, 256, 0, stream>>>(y, feat, centers,
                                                          counts, fsum, out);
    cl_grad_kernel<<<NUM_CLASSES / 8, 256, 0, stream>>>(centers, counts, fsum,
                                                        out + 1);
}